// MultiAgentLinearLayer_63634235458160
// MI455X (gfx1250) — compile-verified
//
#include <hip/hip_runtime.h>
#include <hip/hip_bf16.h>

// ---------------------------------------------------------------------------
// MultiAgentLinearLayer: out[b,:] = W[agent[b]] @ x[b,:] + bias[agent[b]]
// B=2048, I=O=512, 64 agents, fp32.
//
// Bucket batch rows by agent (stable counting sort), then per-agent GEMM with
// V_WMMA_F32_16X16X4_F32 (exact fp32). Each wave register-blocks 2x2 tiles
// (32 batch rows x 32 out cols). K is consumed 8-at-a-time via one b128 load
// per fragment: lanes 0-15 hold k0..k0+3, lanes 16-31 hold k0+4..k0+7, and the
// two WMMAs consume the (permuted) K-slices {k0,k0+1,k0+4,k0+5} (.xy) and
// {k0+2,k0+3,k0+6,k0+7} (.zw). K-order permutation is a valid fp32 sum order.
// Net: 4 b128 loads feed 8 WMMAs per 8-K step (1:2 load:wmma).
// ---------------------------------------------------------------------------

#define NAG        64
#define BATCH_N    2048
#define FEAT       512
#define MAX_STILES 128   // sum_a ceil(count_a/32) <= 2048/32 + 64*(31/32) < 128

typedef __attribute__((ext_vector_type(2))) float v2f;
typedef __attribute__((ext_vector_type(4))) float v4f;
typedef __attribute__((ext_vector_type(8))) float v8f;

// ---------------------------------------------------------------------------
// Kernel 1: stable bucket of batch indices by agent + 32-row super-tile
// descriptors. One block, 64 threads (thread a owns agent a).
// ---------------------------------------------------------------------------
__global__ __launch_bounds__(NAG)
void mal_bucket_kernel(const int* __restrict__ which,
                       int* __restrict__ perm,
                       int* __restrict__ tile_agent,
                       int* __restrict__ tile_start,
                       int* __restrict__ tile_rows)
{
    __shared__ int sc[NAG];
    const int a = threadIdx.x;

    // Invalidate all tile slots (ws is not re-poisoned between calls).
    for (int i = a; i < MAX_STILES; i += NAG) {
        tile_agent[i] = -1;
        tile_start[i] = 0;
        tile_rows[i]  = 0;
    }

    // Pass 1: count my agent's rows.
    int c = 0;
    for (int i = 0; i < BATCH_N; ++i) c += (which[i] == a) ? 1 : 0;
    sc[a] = c;
    __syncthreads();   // counts visible + tile-slot init complete

    // Exclusive prefix sums (element offset and super-tile offset).
    int off = 0, toff = 0;
    for (int j = 0; j < a; ++j) {
        off  += sc[j];
        toff += (sc[j] + 31) >> 5;
    }

    // Pass 2: stable fill of perm (index order => deterministic).
    int idx = off;
    for (int i = 0; i < BATCH_N; ++i) {
        if (which[i] == a) perm[idx++] = i;
    }

    // 32-row super-tile descriptors for this agent.
    const int nt = (c + 31) >> 5;
    for (int t = 0; t < nt; ++t) {
        const int rem = c - 32 * t;
        tile_agent[toff + t] = a;
        tile_start[toff + t] = off + 32 * t;
        tile_rows[toff + t]  = rem > 32 ? 32 : rem;
    }
}

// ---------------------------------------------------------------------------
// Kernel 2: one wave32 per 32(batch) x 32(out) block = 2x2 WMMA tiles.
// D = X_tile[32 x 512] * W[agent]^T[512 x 32] via wmma_f32_16x16x4_f32.
//
// A 16x4 f32 fragment: lane l -> row (l&15), slice-k = 2*(l>>4) + {0,1}
// B 4x16 f32 fragment: lane l -> col (l&15), slice-k = 2*(l>>4) + {0,1}
// C/D 16x16 f32:       vgpr r, lane l -> row r + 8*(l>>4), col (l&15)
// ---------------------------------------------------------------------------
__global__ __launch_bounds__(32)
void mal_wmma_gemm_kernel(const float* __restrict__ X,
                          const float* __restrict__ W,
                          const float* __restrict__ Bias,
                          float* __restrict__ Out,
                          const int* __restrict__ perm,
                          const int* __restrict__ tile_agent,
                          const int* __restrict__ tile_start,
                          const int* __restrict__ tile_rows)
{
    const int tile  = blockIdx.x;
    const int agent = tile_agent[tile];
    if (agent < 0) return;                       // uniform branch: EXEC stays all-1s

    const int start = tile_start[tile];
    const int rows  = tile_rows[tile];           // 1..32
    const int rows0 = rows > 16 ? 16 : rows;     // rows in batch tile 0
    const int rows1 = rows > 16 ? rows - 16 : 0; // rows in batch tile 1
    const int oBase = blockIdx.y * 32;

    const int lane = threadIdx.x;                // 0..31 (wave32)
    const int hh   = lane >> 4;                  // K half selector
    const int lq   = lane & 15;                  // row (A) / col (B,D)

    // Gather X rows for the two A fragments (clamped; clamped D rows never stored).
    const int i0   = start + ((lq < rows0) ? lq : 0);
    const int i1   = (rows1 > 0) ? (start + 16 + ((lq < rows1) ? lq : 0)) : start;
    const int xr0  = perm[i0];
    const int xr1  = perm[i1];

    // Per-lane base: lanes 0-15 read k0..k0+3, lanes 16-31 read k0+4..k0+7.
    const float* a0Ptr = X + (size_t)xr0 * FEAT + 4 * hh;
    const float* a1Ptr = X + (size_t)xr1 * FEAT + 4 * hh;
    const float* wBase = W + (size_t)agent * FEAT * FEAT + 4 * hh;
    const float* b0Ptr = wBase + (size_t)(oBase + lq)      * FEAT;
    const float* b1Ptr = wBase + (size_t)(oBase + 16 + lq) * FEAT;

    v8f acc00 = {}, acc01 = {}, acc10 = {}, acc11 = {};
#pragma unroll 2
    for (int k0 = 0; k0 < FEAT; k0 += 8) {
        v4f a0 = *(const v4f*)(a0Ptr + k0);
        v4f a1 = *(const v4f*)(a1Ptr + k0);
        v4f b0 = *(const v4f*)(b0Ptr + k0);
        v4f b1 = *(const v4f*)(b1Ptr + k0);

        v2f a0lo = __builtin_shufflevector(a0, a0, 0, 1);
        v2f a0hi = __builtin_shufflevector(a0, a0, 2, 3);
        v2f a1lo = __builtin_shufflevector(a1, a1, 0, 1);
        v2f a1hi = __builtin_shufflevector(a1, a1, 2, 3);
        v2f b0lo = __builtin_shufflevector(b0, b0, 0, 1);
        v2f b0hi = __builtin_shufflevector(b0, b0, 2, 3);
        v2f b1lo = __builtin_shufflevector(b1, b1, 0, 1);
        v2f b1hi = __builtin_shufflevector(b1, b1, 2, 3);

        // K-slice {k0, k0+1, k0+4, k0+5}
        acc00 = __builtin_amdgcn_wmma_f32_16x16x4_f32(false, a0lo, false, b0lo, (short)0, acc00, false, false);
        acc01 = __builtin_amdgcn_wmma_f32_16x16x4_f32(false, a0lo, false, b1lo, (short)0, acc01, false, false);
        acc10 = __builtin_amdgcn_wmma_f32_16x16x4_f32(false, a1lo, false, b0lo, (short)0, acc10, false, false);
        acc11 = __builtin_amdgcn_wmma_f32_16x16x4_f32(false, a1lo, false, b1lo, (short)0, acc11, false, false);
        // K-slice {k0+2, k0+3, k0+6, k0+7}
        acc00 = __builtin_amdgcn_wmma_f32_16x16x4_f32(false, a0hi, false, b0hi, (short)0, acc00, false, false);
        acc01 = __builtin_amdgcn_wmma_f32_16x16x4_f32(false, a0hi, false, b1hi, (short)0, acc01, false, false);
        acc10 = __builtin_amdgcn_wmma_f32_16x16x4_f32(false, a1hi, false, b0hi, (short)0, acc10, false, false);
        acc11 = __builtin_amdgcn_wmma_f32_16x16x4_f32(false, a1hi, false, b1hi, (short)0, acc11, false, false);
    }

    // Bias per output column (both column tiles).
    const float bv0 = Bias[(size_t)agent * FEAT + oBase + lq];
    const float bv1 = Bias[(size_t)agent * FEAT + oBase + 16 + lq];

#pragma unroll
    for (int r = 0; r < 8; ++r) {
        const int m = r + 8 * hh;                // local batch row of acc[r]
        if (m < rows0) {
            const int orow = perm[start + m];
            float* p = Out + (size_t)orow * FEAT + oBase + lq;
            p[0]  = acc00[r] + bv0;
            p[16] = acc01[r] + bv1;
        }
        if (m < rows1) {
            const int orow = perm[start + 16 + m];
            float* p = Out + (size_t)orow * FEAT + oBase + lq;
            p[0]  = acc10[r] + bv0;
            p[16] = acc11[r] + bv1;
        }
    }
}

// ---------------------------------------------------------------------------
// Launcher. Inputs (setup_inputs order): which_agents(i32[2048]),
// input(f32[2048,512]), weight(f32[64,512,512]), bias(f32[64,512]).
// Output: f32[2048,512].
// ---------------------------------------------------------------------------
extern "C" void kernel_launch(void* const* d_in, const int* in_sizes, int n_in,
                              void* d_out, int out_size, void* d_ws, size_t ws_size,
                              hipStream_t stream) {
    (void)in_sizes; (void)n_in; (void)out_size; (void)ws_size;

    const int*   which = (const int*)  d_in[0];
    const float* X     = (const float*)d_in[1];
    const float* W     = (const float*)d_in[2];
    const float* Bias  = (const float*)d_in[3];
    float*       Out   = (float*)      d_out;

    int* ws         = (int*)d_ws;
    int* perm       = ws;                         // [2048]
    int* tile_agent = perm + BATCH_N;             // [128]
    int* tile_start = tile_agent + MAX_STILES;    // [128]
    int* tile_rows  = tile_start + MAX_STILES;    // [128]

    mal_bucket_kernel<<<1, NAG, 0, stream>>>(which, perm, tile_agent,
                                             tile_start, tile_rows);

    dim3 grid(MAX_STILES, FEAT / 32);
    mal_wmma_gemm_kernel<<<grid, 32, 0, stream>>>(X, W, Bias, Out, perm,
                                                  tile_agent, tile_start,
                                                  tile_rows);
}